// ContrastiveSingleProsodyLoss_76270029242717
// MI455X (gfx1250) — compile-verified
//
#include <hip/hip_runtime.h>
#include <hip/hip_bf16.h>
#include <math.h>

// ---------------------------------------------------------------------------
// ContrastiveSingleProsodyLoss for gfx1250 (MI455X, wave32).
//
// denom[i] = sum_j exp(2/(1+|r_j - r_i|)) computed with per-wave 16-row tiles;
// the j-reduction runs on the matrix pipe: V_WMMA_F32_16X16X4_F32 with an
// all-ones B matrix (D = A*1 + C accumulates f32 row sums at full precision).
// Hot-loop transcendentals use hardware v_rcp_f32/v_exp_f32 (args are in a
// guard-free range: exp arg in (0,2], divisor >= 1), finalize uses libm.
// ---------------------------------------------------------------------------

typedef __attribute__((ext_vector_type(2))) float v2f;
typedef __attribute__((ext_vector_type(8))) float v8f;

#define B_SZ 8192
#define N_SZ 16384
#define JS   8                 // j-segments (parallelism across the j loop)
#define SEG  (N_SZ / JS)       // 2048 j's per segment -> 8KB LDS tile
#define WAVES_PER_BLOCK 8
#define ROWS_PER_BLOCK  (WAVES_PER_BLOCK * 16)   // 128 i-rows per block

// 2 * log2(e): exp(2*s) == exp2(TWO_LOG2E * s)
#define TWO_LOG2E 2.8853900817779268f

__global__ void csp_zero_ws(float* __restrict__ ws) {
    int idx = blockIdx.x * blockDim.x + threadIdx.x;
    if (idx < N_SZ) ws[idx] = 0.0f;
}

__device__ __forceinline__ float pair_exp(float rj, float ri) {
    // exp(2 / (1 + |rj - ri|)) via hw rcp + hw exp2; arg in (0, 2.886]
    const float d = fabsf(rj - ri);
    return __builtin_amdgcn_exp2f(TWO_LOG2E * __builtin_amdgcn_rcpf(1.0f + d));
}

// grid: (N_SZ / ROWS_PER_BLOCK) * JS = 128 * 8 = 1024 blocks, 256 threads.
__global__ void __launch_bounds__(256)
csp_denom_kernel(const float* __restrict__ emb_i,
                 const float* __restrict__ emb_j,
                 float* __restrict__ ws) {
    __shared__ float repj[SEG];

    const int seg   = blockIdx.x % JS;
    const int iblk  = blockIdx.x / JS;
    const int t     = threadIdx.x;
    const int jbase = seg * SEG;

    // Stage this block's j-segment of rep = concat(emb_i, emb_j) into LDS.
    for (int v = t; v < SEG; v += 256) {
        const int j = jbase + v;
        repj[v] = (j < B_SZ) ? emb_i[j] : emb_j[j - B_SZ];
    }
    __syncthreads();

    const int lane = t & 31;          // wave32
    const int wave = t >> 5;
    const int i0   = iblk * ROWS_PER_BLOCK + wave * 16;

    // A-matrix (16x4 f32) layout: lane L<16 -> M=L, K={0,1}; L>=16 -> M=L-16, K={2,3}
    const int m = lane & 15;
    const int i = i0 + m;
    const float ri  = (i < B_SZ) ? emb_i[i] : emb_j[i - B_SZ];
    const int koff  = (lane >> 4) * 2;   // 0 or 2

    v8f c = {};                           // f32 row-sum accumulator (C/D)
    v2f ones; ones.x = 1.0f; ones.y = 1.0f;  // B = ones(4x16): every element 1.0

#pragma unroll 4
    for (int jb = 0; jb < SEG; jb += 4) {
        v2f a;
        a.x = pair_exp(repj[jb + koff],     ri);
        a.y = pair_exp(repj[jb + koff + 1], ri);
        // D[m][n] = sum_k A[m][k] + C[m][n] -> every column holds the row sum.
        c = __builtin_amdgcn_wmma_f32_16x16x4_f32(
                /*neg_a=*/false, a, /*neg_b=*/false, ones,
                /*c_mod=*/(short)0, c, /*reuse_a=*/false, /*reuse_b=*/false);
    }

    // C/D layout: VGPR r, lane 0  -> (M=r,   N=0);  lane 16 -> (M=8+r, N=0).
    // Lane 0 holds row sums 0..7, lane 16 holds 8..15.
    if ((lane & 15) == 0) {
        const int base = i0 + (lane >> 4) * 8;
#pragma unroll
        for (int r = 0; r < 8; ++r) {
            atomicAdd(&ws[base + r], c[r]);
        }
    }
}

// Single block, 1024 threads: prosody softmax + final log-sum reduction.
// O(B) work: keep full-precision libm here (it sets the final loss digits).
__global__ void __launch_bounds__(1024)
csp_finalize_kernel(const float* __restrict__ emb_i,
                    const float* __restrict__ emb_j,
                    const float* __restrict__ pro_i,
                    const float* __restrict__ pro_j,
                    const float* __restrict__ ws,
                    float* __restrict__ out) {
    __shared__ float red[1024];
    const int t = threadIdx.x;

    // 1) max over prosody_diff (stable softmax, matches jax.nn.softmax)
    float mx = -INFINITY;
    for (int k = t; k < B_SZ; k += 1024) mx = fmaxf(mx, fabsf(pro_i[k] - pro_j[k]));
    red[t] = mx; __syncthreads();
    for (int s = 512; s > 0; s >>= 1) {
        if (t < s) red[t] = fmaxf(red[t], red[t + s]);
        __syncthreads();
    }
    mx = red[0]; __syncthreads();

    // 2) softmax denominator
    float sme = 0.0f;
    for (int k = t; k < B_SZ; k += 1024) sme += expf(fabsf(pro_i[k] - pro_j[k]) - mx);
    red[t] = sme; __syncthreads();
    for (int s = 512; s > 0; s >>= 1) {
        if (t < s) red[t] += red[t + s];
        __syncthreads();
    }
    const float S = red[0]; __syncthreads();

    // 3) loss = (1/N)[ sum_i log(denom_i) + 2*sum_k log(sm_k+eps) + 2*sum_k log(1+|di-dj|) ]
    const float E2 = expf(2.0f);   // diagonal term exp(sim_ii/T), sim_ii == 1
    float acc = 0.0f;
    for (int i = t; i < N_SZ; i += 1024) {
        acc += logf(ws[i] - E2);
    }
    for (int k = t; k < B_SZ; k += 1024) {
        const float sm_k = expf(fabsf(pro_i[k] - pro_j[k]) - mx) / S;
        const float d    = fabsf(emb_i[k] - emb_j[k]);
        acc += 2.0f * (logf(sm_k + 0.01f) + logf(1.0f + d));
    }
    red[t] = acc; __syncthreads();
    for (int s = 512; s > 0; s >>= 1) {
        if (t < s) red[t] += red[t + s];
        __syncthreads();
    }
    if (t == 0) out[0] = red[0] / (float)N_SZ;
}

extern "C" void kernel_launch(void* const* d_in, const int* in_sizes, int n_in,
                              void* d_out, int out_size, void* d_ws, size_t ws_size,
                              hipStream_t stream) {
    const float* emb_i = (const float*)d_in[0];   // [B,1] -> B floats
    const float* emb_j = (const float*)d_in[1];   // [B,1] -> B floats
    const float* pro_i = (const float*)d_in[2];   // [B]
    const float* pro_j = (const float*)d_in[3];   // [B]
    float* out = (float*)d_out;                   // scalar loss
    float* ws  = (float*)d_ws;                    // N_SZ floats of denom accumulators

    csp_zero_ws<<<N_SZ / 256, 256, 0, stream>>>(ws);

    const int denom_blocks = (N_SZ / ROWS_PER_BLOCK) * JS;   // 1024
    csp_denom_kernel<<<denom_blocks, 256, 0, stream>>>(emb_i, emb_j, ws);

    csp_finalize_kernel<<<1, 1024, 0, stream>>>(emb_i, emb_j, pro_i, pro_j, ws, out);
}